// ScaledDotProductAttention_24154896073233
// MI455X (gfx1250) — compile-verified
//
#include <hip/hip_runtime.h>

typedef __attribute__((ext_vector_type(2))) float v2f;
typedef __attribute__((ext_vector_type(8))) float v8f;

#define S_LEN  2048
#define D_HEAD 64
#define NKBLK  (S_LEN / 16)   // 128 key blocks of 16

// One wave (32 lanes) handles one 16-query block of one (b,h) head.
// Flash-attention single pass; raw scaled scores streamed out non-temporally.
__global__ __launch_bounds__(32)
void sdpa_fwd_kernel(const float* __restrict__ Q,
                     const float* __restrict__ K,
                     const float* __restrict__ V,
                     float* __restrict__ Out,
                     float* __restrict__ Scores)
{
    __shared__ float pTile[16 * 16];   // P tile bounce buffer (C/D layout -> A layout)

    const int lane = threadIdx.x & 31;
    const int m    = lane & 15;   // A: row M ; B/C/D: column N
    const int h    = lane >> 4;   // half select: K-group / row-high-half

    const int blk  = blockIdx.x;
    const int qblk = blk & (NKBLK - 1);   // 128 query blocks per head
    const int bh   = blk >> 7;            // batch*head in [0,32)
    const int q0   = qblk * 16;

    const float* Qh = Q + ((size_t)bh * S_LEN + q0) * D_HEAD;
    const float* Kh = K + (size_t)bh * S_LEN * D_HEAD;
    const float* Vh = V + (size_t)bh * S_LEN * D_HEAD;

    // ---- Q block (16x64) as 16 fp32 A-tiles (16x4 each) ----
    // A layout: lane(0-15)=M, v0 = K(4t+2h), v1 = K(4t+2h+1)  -> one float2 load
    v2f qa[16];
#pragma unroll
    for (int t = 0; t < 16; ++t) {
        qa[t] = *(const v2f*)(Qh + (size_t)m * D_HEAD + 4 * t + 2 * h);
    }

    v8f acc[4];
#pragma unroll
    for (int n = 0; n < 4; ++n) { v8f z = {}; acc[n] = z; }

    float Mrun[8], Lrun[8];
#pragma unroll
    for (int r = 0; r < 8; ++r) { Mrun[r] = -__builtin_inff(); Lrun[r] = 0.0f; }

    const float scale = 0.125f;  // 1/sqrt(64)

    for (int j = 0; j < NKBLK; ++j) {
        const int j0 = j * 16;

        if (j + 1 < NKBLK) {  // prefetch next K/V blocks (global_prefetch_b8)
            __builtin_prefetch(Kh + (size_t)(j0 + 16 + m) * D_HEAD + 32 * h, 0, 0);
            __builtin_prefetch(Vh + (size_t)(j0 + 16 + m) * D_HEAD + 32 * h, 0, 0);
        }

        // ---- S tile = Q x K^T : K block as 16 B-tiles (4x16), same float2 pattern ----
        v8f c = {};
#pragma unroll
        for (int t = 0; t < 16; ++t) {
            v2f kb = *(const v2f*)(Kh + (size_t)(j0 + m) * D_HEAD + 4 * t + 2 * h);
            c = __builtin_amdgcn_wmma_f32_16x16x4_f32(false, qa[t], false, kb,
                                                      (short)0, c, false, false);
        }
#pragma unroll
        for (int r = 0; r < 8; ++r) c[r] *= scale;

        // ---- stream raw scaled scores (written once, never re-read: NT store) ----
        {
            float* sp = Scores + ((size_t)bh * S_LEN + q0) * S_LEN + j0;
#pragma unroll
            for (int r = 0; r < 8; ++r) {
                const int row = r + 8 * h;            // C/D layout: vgpr r, half h
                __builtin_nontemporal_store(c[r], sp + (size_t)row * S_LEN + m);
            }
        }

        // ---- online softmax (rows live inside 16-lane halves; xor 1/2/4/8) ----
        float alpha[8];
        v8f p8;
#pragma unroll
        for (int r = 0; r < 8; ++r) {
            float mx = c[r];
            mx = fmaxf(mx, __shfl_xor(mx, 1, 32));
            mx = fmaxf(mx, __shfl_xor(mx, 2, 32));
            mx = fmaxf(mx, __shfl_xor(mx, 4, 32));
            mx = fmaxf(mx, __shfl_xor(mx, 8, 32));
            const float Mnew = fmaxf(Mrun[r], mx);
            alpha[r] = __expf(Mrun[r] - Mnew);
            const float pv = __expf(c[r] - Mnew);
            float sum = pv;
            sum += __shfl_xor(sum, 1, 32);
            sum += __shfl_xor(sum, 2, 32);
            sum += __shfl_xor(sum, 4, 32);
            sum += __shfl_xor(sum, 8, 32);
            Lrun[r] = Lrun[r] * alpha[r] + sum;
            Mrun[r] = Mnew;
            p8[r]   = pv;
        }

        // rescale running output accumulator
#pragma unroll
        for (int n = 0; n < 4; ++n)
#pragma unroll
            for (int r = 0; r < 8; ++r)
                acc[n][r] *= alpha[r];

        // ---- P tile: C/D layout -> LDS row-major (DS ops in-order within wave) ----
#pragma unroll
        for (int r = 0; r < 8; ++r) {
            pTile[(r + 8 * h) * 16 + m] = p8[r];
        }

        // ---- O += P x V : reload P as 4 A-tiles, V block as 16 B-tiles ----
#pragma unroll
        for (int t = 0; t < 4; ++t) {
            v2f pa = *(const v2f*)&pTile[m * 16 + 4 * t + 2 * h];
            const float* v0p = Vh + (size_t)(j0 + 4 * t + 2 * h) * D_HEAD;
#pragma unroll
            for (int n = 0; n < 4; ++n) {
                v2f bv;
                bv[0] = v0p[16 * n + m];            // k = 4t+2h   row of V
                bv[1] = v0p[D_HEAD + 16 * n + m];   // k = 4t+2h+1 row of V
                acc[n] = __builtin_amdgcn_wmma_f32_16x16x4_f32(false, pa, false, bv,
                                                               (short)0, acc[n],
                                                               false, false);
            }
        }
    }

    // ---- finalize: divide by row sums, store output ----
    float* op = Out + ((size_t)bh * S_LEN + q0) * D_HEAD;
#pragma unroll
    for (int r = 0; r < 8; ++r) {
        const int row = r + 8 * h;
        const float inv = 1.0f / Lrun[r];
#pragma unroll
        for (int n = 0; n < 4; ++n) {
            __builtin_nontemporal_store(acc[n][r] * inv,
                                        op + (size_t)row * D_HEAD + 16 * n + m);
        }
    }
}

extern "C" void kernel_launch(void* const* d_in, const int* in_sizes, int n_in,
                              void* d_out, int out_size, void* d_ws, size_t ws_size,
                              hipStream_t stream) {
    (void)in_sizes; (void)n_in; (void)out_size; (void)d_ws; (void)ws_size;
    const float* Q = (const float*)d_in[0];
    const float* K = (const float*)d_in[1];
    const float* V = (const float*)d_in[2];

    const int B = 2, H = 16;
    float* Out    = (float*)d_out;                                  // [B,H,S,D]
    float* Scores = Out + (size_t)B * H * S_LEN * D_HEAD;           // [B,H,S,S]

    dim3 grid(B * H * (S_LEN / 16));   // 4096 waves, one per 16-query block
    dim3 block(32);                    // one wave32
    sdpa_fwd_kernel<<<grid, block, 0, stream>>>(Q, K, V, Out, Scores);
}